// ConstellationNet_4398046511426
// MI455X (gfx1250) — compile-verified
//
#include <hip/hip_runtime.h>
#include <math.h>

typedef __attribute__((ext_vector_type(2))) float v2f;
typedef __attribute__((ext_vector_type(8))) float v8f;

__device__ __forceinline__ v8f wmma_f32x4(v2f a, v2f b, v8f c) {
  // D = A(16x4) * B(4x16) + C, fp32, wave32
  return __builtin_amdgcn_wmma_f32_16x16x4_f32(false, a, false, b, (short)0, c, false, false);
}

// -------------------------------------------------------------------------
// utils
// -------------------------------------------------------------------------
__global__ void zero_kernel(float* __restrict__ p, int n) {
  int i = blockIdx.x * blockDim.x + threadIdx.x;
  if (i < n) p[i] = 0.f;
}

// conv 3x3, pad 1, 3 input channels, weights OIHW (64,3,3,3)
__device__ __forceinline__ float conv3x3(const float* __restrict__ X,
                                         const float* __restrict__ W,
                                         const float* __restrict__ bias,
                                         int b, int oc, int y, int x, int H) {
  float acc = bias[oc];
#pragma unroll
  for (int ic = 0; ic < 3; ++ic) {
    const float* Xc = X + ((size_t)(b * 3 + ic) * H) * H;
    const float* Wc = W + (oc * 3 + ic) * 9;
#pragma unroll
    for (int ky = 0; ky < 3; ++ky) {
      int yy = y + ky - 1;
      if (yy < 0 || yy >= H) continue;
#pragma unroll
      for (int kx = 0; kx < 3; ++kx) {
        int xx = x + kx - 1;
        if (xx < 0 || xx >= H) continue;
        acc += Xc[yy * H + xx] * Wc[ky * 3 + kx];
      }
    }
  }
  return acc;
}

// -------------------------------------------------------------------------
// conv4 path: stats pass then fused BN+ReLU+maxpool (conv recomputed)
// -------------------------------------------------------------------------
__global__ __launch_bounds__(256) void conv4_stats_kernel(
    const float* __restrict__ X, const float* __restrict__ W,
    const float* __restrict__ bias, float* __restrict__ sum,
    float* __restrict__ sumsq, int H) {
  int oc = blockIdx.y;
  int idx = blockIdx.x * 256 + threadIdx.x;
  int total = 16 * H * H;
  float v = 0.f;
  if (idx < total) {
    int b = idx / (H * H);
    int rem = idx % (H * H);
    v = conv3x3(X, W, bias, b, oc, rem / H, rem % H, H);
  }
  __shared__ float s1[256], s2[256];
  s1[threadIdx.x] = v;
  s2[threadIdx.x] = v * v;
  __syncthreads();
  for (int st = 128; st > 0; st >>= 1) {
    if (threadIdx.x < st) {
      s1[threadIdx.x] += s1[threadIdx.x + st];
      s2[threadIdx.x] += s2[threadIdx.x + st];
    }
    __syncthreads();
  }
  if (threadIdx.x == 0) {
    atomicAdd(&sum[oc], s1[0]);
    atomicAdd(&sumsq[oc], s2[0]);
  }
}

__global__ __launch_bounds__(256) void conv4_cfm_kernel(
    const float* __restrict__ X, const float* __restrict__ W,
    const float* __restrict__ bias, const float* __restrict__ sum,
    const float* __restrict__ sumsq, float* __restrict__ cfm, int H, int h) {
  int idx = blockIdx.x * 256 + threadIdx.x;
  int total = 16 * 64 * h * h;
  if (idx >= total) return;
  int x = idx % h;
  int t = idx / h;
  int y = t % h;
  t /= h;
  int oc = t % 64;
  int b = t / 64;
  float cnt = 16.f * (float)H * (float)H;
  float mean = sum[oc] / cnt;
  float var = sumsq[oc] / cnt - mean * mean;
  float inv = rsqrtf(var + 1e-5f);
  float best = -INFINITY;
#pragma unroll
  for (int dy = 0; dy < 2; ++dy)
#pragma unroll
    for (int dx = 0; dx < 2; ++dx) {
      float v = conv3x3(X, W, bias, b, oc, 2 * y + dy, 2 * x + dx, H);
      v = fmaxf((v - mean) * inv, 0.f);
      best = fmaxf(best, v);
    }
  cfm[idx] = best;
}

// -------------------------------------------------------------------------
// clustering
// -------------------------------------------------------------------------
__global__ __launch_bounds__(256) void cluster_setup_kernel(
    const float* __restrict__ v0, float* __restrict__ v, float* __restrict__ vsq,
    float* __restrict__ s, float* __restrict__ delta) {
  for (int i = threadIdx.x; i < 2048; i += 256) v[i] = v0[i];
  __syncthreads();
  if (threadIdx.x < 32) {
    float acc = 0.f;
    for (int c = 0; c < 64; ++c) {
      float t = v[threadIdx.x * 64 + c];
      acc += t * t;
    }
    vsq[threadIdx.x] = acc;
    s[threadIdx.x] = 0.f;
    delta[threadIdx.x] = 0.f;
  }
}

// vp[j] = sum over all (b,pos) of cfm[b][j][pos]  (reference bug preserved:
// each row of v_p is the scalar channel-sum broadcast across 64 dims)
__global__ __launch_bounds__(256) void vp_kernel(const float* __restrict__ cfm,
                                                 float* __restrict__ vp, int hw) {
  int j = blockIdx.x;  // 0..31
  float acc = 0.f;
  int total = 16 * hw;
  for (int t = threadIdx.x; t < total; t += 256) {
    int b = t / hw, pos = t % hw;
    acc += cfm[((size_t)(b * 64 + j)) * hw + pos];
  }
  __shared__ float sm[256];
  sm[threadIdx.x] = acc;
  __syncthreads();
  for (int st = 128; st > 0; st >>= 1) {
    if (threadIdx.x < st) sm[threadIdx.x] += sm[threadIdx.x + st];
    __syncthreads();
  }
  if (threadIdx.x == 0) vp[j] = sm[0];
}

__global__ __launch_bounds__(256) void p2_kernel(const float* __restrict__ cfm,
                                                 float* __restrict__ p2, int hw,
                                                 int N) {
  int n = blockIdx.x * 256 + threadIdx.x;
  if (n >= N) return;
  int b = n / hw, pos = n % hw;
  const float* base = cfm + (size_t)b * 64 * hw + pos;
  float acc = 0.f;
  for (int c = 0; c < 64; ++c) {
    float t = base[(size_t)c * hw];
    acc += t * t;
  }
  p2[n] = acc;
}

// One wave per 16 points: dot = pts(16x64) @ v^T(64x32) via 2x16 chained
// V_WMMA_F32_16X16X4_F32; then euclidean distance, row softmax over the 32
// clusters (beta=100), accumulate per-cluster membership mass.
__global__ __launch_bounds__(32) void cluster_dist_kernel(
    const float* __restrict__ cfm, const float* __restrict__ v,
    const float* __restrict__ p2, const float* __restrict__ vsq,
    float* __restrict__ delta, float* __restrict__ dbuf, int h, int N,
    int write_d, float beta) {
  int tile = blockIdx.x;
  int lane = threadIdx.x;
  int col = lane & 15;
  int hi = lane >> 4;
  int hh = h * h;
  int arow = tile * 16 + col;  // A-matrix row held by this lane

  float ax[16], ay[16];
  if (arow < N) {
    int b = arow / hh, rem = arow % hh;
    const float* base = cfm + (size_t)b * 64 * hh + rem;  // channel stride hh
#pragma unroll
    for (int c = 0; c < 16; ++c) {
      int k0 = 4 * c + 2 * hi;
      ax[c] = base[(size_t)k0 * hh];
      ay[c] = base[(size_t)(k0 + 1) * hh];
    }
  } else {
#pragma unroll
    for (int c = 0; c < 16; ++c) {
      ax[c] = 0.f;
      ay[c] = 0.f;
    }
  }

  float dl[2][8];
#pragma unroll
  for (int jt = 0; jt < 2; ++jt) {
    int j = jt * 16 + col;
    v8f acc = {};
#pragma unroll
    for (int c = 0; c < 16; ++c) {
      int k0 = 4 * c + 2 * hi;
      v2f a, bb;
      a[0] = ax[c];
      a[1] = ay[c];
      bb[0] = v[j * 64 + k0];
      bb[1] = v[j * 64 + k0 + 1];
      acc = wmma_f32x4(a, bb, acc);
    }
    float vs = vsq[j];
#pragma unroll
    for (int r = 0; r < 8; ++r) {
      int grow = tile * 16 + r + 8 * hi;
      float pp = (grow < N) ? p2[grow] : 0.f;
      float d2 = pp + vs - 2.f * acc[r];
      dl[jt][r] = sqrtf(fmaxf(d2, 0.f));
    }
  }

  float dacc0 = 0.f, dacc1 = 0.f;
#pragma unroll
  for (int r = 0; r < 8; ++r) {
    int grow = tile * 16 + r + 8 * hi;
    bool rv = grow < N;
    float l0 = -beta * dl[0][r];
    float l1 = -beta * dl[1][r];
    float lm = fmaxf(l0, l1);
    for (int m = 1; m <= 8; m <<= 1) lm = fmaxf(lm, __shfl_xor(lm, m, 32));
    float e0 = expf(l0 - lm), e1 = expf(l1 - lm);
    float ssum = e0 + e1;
    for (int m = 1; m <= 8; m <<= 1) ssum += __shfl_xor(ssum, m, 32);
    float inv = 1.f / ssum;
    if (rv) {
      dacc0 += e0 * inv;
      dacc1 += e1 * inv;
      if (write_d) {
        dbuf[(size_t)grow * 32 + col] = dl[0][r];
        dbuf[(size_t)grow * 32 + 16 + col] = dl[1][r];
      }
    }
  }
  atomicAdd(&delta[col], dacc0);
  atomicAdd(&delta[16 + col], dacc1);
}

__global__ void cluster_update_kernel(float* __restrict__ v, float* __restrict__ s,
                                      float* __restrict__ delta,
                                      const float* __restrict__ vp,
                                      float* __restrict__ vsq, float lbda) {
  int j = threadIdx.x;
  if (j >= 32) return;
  float del = delta[j];
  float mu = lbda / (s[j] + del);
  float vpj = vp[j];
  float acc = 0.f;
  for (int c = 0; c < 64; ++c) {
    float nv = (1.f - mu) * v[j * 64 + c] + mu * vpj;
    v[j * 64 + c] = nv;
    acc += nv * nv;
  }
  vsq[j] = acc;
  s[j] += del;
  delta[j] = 0.f;
}

// -------------------------------------------------------------------------
// positional encoding: pe[c][y][x], c<16 keyed by y, c>=16 keyed by x
// -------------------------------------------------------------------------
__global__ __launch_bounds__(256) void pe_kernel(float* __restrict__ pe, int h) {
  int idx = blockIdx.x * 256 + threadIdx.x;
  int total = 32 * h * h;
  if (idx >= total) return;
  int x = idx % h;
  int y = (idx / h) % h;
  int c = idx / (h * h);
  int p = (c < 16) ? y : x;
  int t = c & 15;
  float div = expf(-logf(10000.f) / 16.f * (float)(2 * (t & 7)));
  float arg = (float)p * div;
  pe[idx] = (t < 8) ? sinf(arg) : cosf(arg);
}

// -------------------------------------------------------------------------
// attention: Q/K from (d_map + pe), V from d_map; rows >= hw zero-padded
// -------------------------------------------------------------------------
__global__ __launch_bounds__(256) void qkv_kernel(
    const float* __restrict__ dbuf, const float* __restrict__ pe,
    const float* __restrict__ Wq, const float* __restrict__ bq,
    const float* __restrict__ Wk, const float* __restrict__ bk,
    const float* __restrict__ Wv, const float* __restrict__ bv,
    float* __restrict__ Q, float* __restrict__ K, float* __restrict__ V, int h,
    int hw, int hwp) {
  int bh = blockIdx.y;
  int b = bh >> 2;
  int hd = bh & 3;
  int idx = blockIdx.x * 256 + threadIdx.x;
  if (idx >= hwp * 32) return;
  int i = idx >> 5;
  int c = idx & 31;
  size_t o = ((size_t)bh * hwp + i) * 32 + c;
  if (i >= hw) {
    Q[o] = 0.f;
    K[o] = 0.f;
    V[o] = 0.f;
    return;
  }
  int y = i / h, x = i % h;
  int n = (b * h + y) * h + x;
  const float* drow = dbuf + (size_t)n * 32;
  const float* wq = Wq + ((size_t)hd * 32 + c) * 32;
  const float* wk = Wk + ((size_t)hd * 32 + c) * 32;
  const float* wv = Wv + ((size_t)hd * 32 + c) * 32;
  float aq = bq[hd * 32 + c], ak = bk[hd * 32 + c], av = bv[hd * 32 + c];
  for (int k = 0; k < 32; ++k) {
    float dv = drow[k];
    float f1 = dv + pe[(k * h + y) * h + x];
    aq += f1 * wq[k];
    ak += f1 * wk[k];
    av += dv * wv[k];
  }
  Q[o] = aq;
  K[o] = ak;
  V[o] = av;
}

// Flash-style streaming scores: per row i keep only Z_i = sum_j exp(s-m).
// max_j softmax(S)_ij == 1/Z_i, which is all the reference uses.
// One wave per 16-row tile; 16x16 score tiles via 8 chained fp32 WMMAs.
__global__ __launch_bounds__(32) void attn_scores_kernel(
    const float* __restrict__ Q, const float* __restrict__ Km,
    float* __restrict__ Zbuf, int hw, int hwp) {
  int bh = blockIdx.y;
  int rt = blockIdx.x;
  int lane = threadIdx.x;
  int col = lane & 15;
  int hi = lane >> 4;
  const float* Qb = Q + (size_t)bh * hwp * 32;
  const float* Kb = Km + (size_t)bh * hwp * 32;

  int arow = rt * 16 + col;
  float ax[8], ay[8];
#pragma unroll
  for (int c = 0; c < 8; ++c) {
    int k0 = 4 * c + 2 * hi;
    ax[c] = Qb[arow * 32 + k0];
    ay[c] = Qb[arow * 32 + k0 + 1];
  }
  float M[8], Z[8];
#pragma unroll
  for (int r = 0; r < 8; ++r) {
    M[r] = -INFINITY;
    Z[r] = 0.f;
  }
  const float scale = 0.17677669529663687f;  // 1/sqrt(32)
  int ntiles = hwp >> 4;
  for (int jt = 0; jt < ntiles; ++jt) {
    int jrow = jt * 16 + col;
    v8f acc = {};
#pragma unroll
    for (int c = 0; c < 8; ++c) {
      int k0 = 4 * c + 2 * hi;
      v2f a, bb;
      a[0] = ax[c];
      a[1] = ay[c];
      bb[0] = Kb[jrow * 32 + k0];
      bb[1] = Kb[jrow * 32 + k0 + 1];
      acc = wmma_f32x4(a, bb, acc);
    }
    bool cv = jrow < hw;
#pragma unroll
    for (int r = 0; r < 8; ++r) {
      float s = cv ? acc[r] * scale : -INFINITY;
      float tm = s;
      for (int m = 1; m <= 8; m <<= 1) tm = fmaxf(tm, __shfl_xor(tm, m, 32));
      float nM = fmaxf(M[r], tm);
      float p = cv ? expf(s - nM) : 0.f;
      float ts = p;
      for (int m = 1; m <= 8; m <<= 1) ts += __shfl_xor(ts, m, 32);
      Z[r] = Z[r] * expf(M[r] - nM) + ts;
      M[r] = nM;
    }
  }
  if (col == 0) {
#pragma unroll
    for (int r = 0; r < 8; ++r) {
      int row = rt * 16 + r + 8 * hi;
      if (row < hw) Zbuf[(size_t)bh * hwp + row] = Z[r];
    }
  }
}

// fa[b][i][c] = bo[c] + sum_hd (1/Z) * sum_k Wo[c][hd*32+k] * V[b,hd,i,k]
__global__ __launch_bounds__(256) void attn_out_kernel(
    const float* __restrict__ V, const float* __restrict__ Z,
    const float* __restrict__ Wo, const float* __restrict__ bo,
    float* __restrict__ fa, int hw, int hwp) {
  int b = blockIdx.y;
  int idx = blockIdx.x * 256 + threadIdx.x;
  if (idx >= hw * 32) return;
  int i = idx >> 5;
  int c = idx & 31;
  float acc = bo[c];
#pragma unroll
  for (int hd = 0; hd < 4; ++hd) {
    float invz = 1.f / Z[(size_t)(b * 4 + hd) * hwp + i];
    const float* vrow = V + ((size_t)(b * 4 + hd) * hwp + i) * 32;
    const float* wrow = Wo + (size_t)c * 128 + hd * 32;
    float a2 = 0.f;
    for (int k = 0; k < 32; ++k) a2 += wrow[k] * vrow[k];
    acc += a2 * invz;
  }
  fa[((size_t)b * hw + i) * 32 + c] = acc;
}

// -------------------------------------------------------------------------
// conv1 (1x1, concat[fa(32), cfm(64)] -> 3), stats, BN+ReLU
// -------------------------------------------------------------------------
__global__ __launch_bounds__(256) void conv1_kernel(
    const float* __restrict__ fa, const float* __restrict__ cfm,
    const float* __restrict__ W, const float* __restrict__ bias,
    float* __restrict__ t3, int hw) {
  int idx = blockIdx.x * 256 + threadIdx.x;
  int total = 16 * 3 * hw;
  if (idx >= total) return;
  int pos = idx % hw;
  int o = (idx / hw) % 3;
  int b = idx / (hw * 3);
  float acc = bias[o];
  const float* w = W + o * 96;
  const float* farow = fa + ((size_t)b * hw + pos) * 32;
  for (int c = 0; c < 32; ++c) acc += w[c] * farow[c];
  const float* cbase = cfm + (size_t)b * 64 * hw + pos;
  for (int c = 0; c < 64; ++c) acc += w[32 + c] * cbase[(size_t)c * hw];
  t3[idx] = acc;
}

__global__ __launch_bounds__(256) void stats_nchw_kernel(
    const float* __restrict__ t, float* __restrict__ sum,
    float* __restrict__ sumsq, int C, int hw) {
  int ch = blockIdx.y;
  int idx = blockIdx.x * 256 + threadIdx.x;
  int total = 16 * hw;
  float v = 0.f;
  if (idx < total) {
    int b = idx / hw, pos = idx % hw;
    v = t[((size_t)(b * C + ch)) * hw + pos];
  }
  __shared__ float s1[256], s2[256];
  s1[threadIdx.x] = v;
  s2[threadIdx.x] = v * v;
  __syncthreads();
  for (int st = 128; st > 0; st >>= 1) {
    if (threadIdx.x < st) {
      s1[threadIdx.x] += s1[threadIdx.x + st];
      s2[threadIdx.x] += s2[threadIdx.x + st];
    }
    __syncthreads();
  }
  if (threadIdx.x == 0) {
    atomicAdd(&sum[ch], s1[0]);
    atomicAdd(&sumsq[ch], s2[0]);
  }
}

__global__ __launch_bounds__(256) void bn_relu_kernel(
    const float* __restrict__ t, const float* __restrict__ sum,
    const float* __restrict__ sumsq, float* __restrict__ out, int C, int hw) {
  int idx = blockIdx.x * 256 + threadIdx.x;
  int total = 16 * C * hw;
  if (idx >= total) return;
  int ch = (idx / hw) % C;
  float cnt = 16.f * (float)hw;
  float mean = sum[ch] / cnt;
  float var = sumsq[ch] / cnt - mean * mean;
  out[idx] = fmaxf((t[idx] - mean) * rsqrtf(var + 1e-5f), 0.f);
}

__global__ void avgpool_kernel(const float* __restrict__ X, float* __restrict__ out) {
  int idx = threadIdx.x;  // 192 = 16*3*2*2
  if (idx >= 192) return;
  int x = idx % 2;
  int y = (idx / 2) % 2;
  int o = (idx / 4) % 3;
  int b = idx / 12;
  const float* base = X + (((size_t)(b * 3 + o) * 5) + 2 * y) * 5 + 2 * x;
  out[idx] = 0.25f * (base[0] + base[1] + base[5] + base[6]);
}

// -------------------------------------------------------------------------
// host orchestration
// -------------------------------------------------------------------------
extern "C" void kernel_launch(void* const* d_in, const int* in_sizes, int n_in,
                              void* d_out, int out_size, void* d_ws, size_t ws_size,
                              hipStream_t stream) {
  (void)in_sizes; (void)n_in; (void)out_size; (void)ws_size;
  const float* X_in    = (const float*)d_in[0];
  // d_in[1] = archi (int scalar) : reference takes archi==0 path
  const float* conv4_w = (const float*)d_in[2];
  const float* conv4_b = (const float*)d_in[3];
  const float* conv1_w = (const float*)d_in[4];
  const float* conv1_b = (const float*)d_in[5];
  const float* Wq      = (const float*)d_in[6];
  const float* bq      = (const float*)d_in[7];
  const float* Wk      = (const float*)d_in[8];
  const float* bk      = (const float*)d_in[9];
  const float* Wv      = (const float*)d_in[10];
  const float* bv      = (const float*)d_in[11];
  const float* Wo      = (const float*)d_in[12];
  const float* bo      = (const float*)d_in[13];
  const float* v_inits = (const float*)d_in[14];

  float* ws = (float*)d_ws;
  size_t off = 0;
  auto alloc = [&](size_t nf) { float* p = ws + off; off += nf; return p; };
  float* bufA  = alloc((size_t)16 * 3 * 84 * 84);
  float* bufB  = alloc((size_t)16 * 3 * 84 * 84);
  float* sum4  = alloc(64);
  float* sq4   = alloc(64);
  float* cfm   = alloc((size_t)16 * 64 * 42 * 42);
  float* vbuf  = alloc(32 * 64);
  float* vp    = alloc(32);
  float* sbuf  = alloc(32);
  float* delta = alloc(32);
  float* vsq   = alloc(32);
  float* p2    = alloc((size_t)16 * 42 * 42);
  float* dbuf  = alloc((size_t)16 * 42 * 42 * 32);
  float* pebuf = alloc((size_t)32 * 42 * 42);
  float* Qb    = alloc((size_t)16 * 4 * 1776 * 32);
  float* Kb    = alloc((size_t)16 * 4 * 1776 * 32);
  float* Vb    = alloc((size_t)16 * 4 * 1776 * 32);
  float* Zb    = alloc((size_t)16 * 4 * 1776);
  float* fab   = alloc((size_t)16 * 1776 * 32);
  float* t3    = alloc((size_t)16 * 3 * 42 * 42);
  float* sum1  = alloc(4);
  float* sq1   = alloc(4);

  auto cdiv = [](int a, int b) { return (a + b - 1) / b; };
  const int Harr[4] = {84, 42, 21, 10};
  const float* Xc = X_in;
  float* Xout[4] = {bufA, bufB, bufA, bufB};

  for (int it = 0; it < 4; ++it) {
    int H = Harr[it];
    int h = H / 2;
    int hw = h * h;
    int hwp = (hw + 15) & ~15;
    int N = 16 * hw;

    // conv4 + BN + ReLU + maxpool -> cfm
    zero_kernel<<<1, 128, 0, stream>>>(sum4, 128);  // sum4 + sq4 contiguous
    conv4_stats_kernel<<<dim3(cdiv(16 * H * H, 256), 64), 256, 0, stream>>>(
        Xc, conv4_w, conv4_b, sum4, sq4, H);
    conv4_cfm_kernel<<<cdiv(16 * 64 * hw, 256), 256, 0, stream>>>(
        Xc, conv4_w, conv4_b, sum4, sq4, cfm, H, h);

    // soft k-means clustering (10 epochs), last epoch writes distances
    cluster_setup_kernel<<<1, 256, 0, stream>>>(v_inits + (size_t)it * 2048, vbuf,
                                                vsq, sbuf, delta);
    vp_kernel<<<32, 256, 0, stream>>>(cfm, vp, hw);
    p2_kernel<<<cdiv(N, 256), 256, 0, stream>>>(cfm, p2, hw, N);
    for (int ep = 0; ep < 10; ++ep) {
      cluster_dist_kernel<<<N / 16, 32, 0, stream>>>(
          cfm, vbuf, p2, vsq, delta, dbuf, h, N, (ep == 9) ? 1 : 0, 100.f);
      if (ep < 9)
        cluster_update_kernel<<<1, 32, 0, stream>>>(vbuf, sbuf, delta, vp, vsq, 1.0f);
    }

    // positional encoding + QKV projections
    pe_kernel<<<cdiv(32 * hw, 256), 256, 0, stream>>>(pebuf, h);
    qkv_kernel<<<dim3(cdiv(hwp * 32, 256), 64), 256, 0, stream>>>(
        dbuf, pebuf, Wq + (size_t)it * 4096, bq + (size_t)it * 128,
        Wk + (size_t)it * 4096, bk + (size_t)it * 128, Wv + (size_t)it * 4096,
        bv + (size_t)it * 128, Qb, Kb, Vb, h, hw, hwp);

    // streaming attention row-sums (WMMA) + output projection
    attn_scores_kernel<<<dim3(hwp / 16, 64), 32, 0, stream>>>(Qb, Kb, Zb, hw, hwp);
    attn_out_kernel<<<dim3(cdiv(hw * 32, 256), 16), 256, 0, stream>>>(
        Vb, Zb, Wo + (size_t)it * 4096, bo + (size_t)it * 32, fab, hw, hwp);

    // conv1 (1x1) + BN + ReLU -> next X
    conv1_kernel<<<cdiv(16 * 3 * hw, 256), 256, 0, stream>>>(fab, cfm, conv1_w,
                                                             conv1_b, t3, hw);
    zero_kernel<<<1, 8, 0, stream>>>(sum1, 8);  // sum1 + sq1 contiguous
    stats_nchw_kernel<<<dim3(cdiv(16 * hw, 256), 3), 256, 0, stream>>>(t3, sum1, sq1,
                                                                       3, hw);
    bn_relu_kernel<<<cdiv(16 * 3 * hw, 256), 256, 0, stream>>>(t3, sum1, sq1,
                                                               Xout[it], 3, hw);
    Xc = Xout[it];
  }

  avgpool_kernel<<<1, 192, 0, stream>>>(Xc, (float*)d_out);
}